// PixelContrastLoss_47519518162964
// MI455X (gfx1250) — compile-verified
//
#include <hip/hip_runtime.h>
#include <hip/hip_bf16.h>

// ---------------------------------------------------------------------------
// Supervised pixel-contrastive loss, CDNA5 (gfx1250) implementation.
//
// G = (C C^T)/T with C = feats.reshape(8192, 64) (row-permutation of the
// reference's view-major stacking; all row-wise stats are permutation
// invariant, so natural layout is used -> contiguous loads).
//
// Per row r:  m_r   = max_n G[r,n]
//             s_r   = sum_{label!=} exp(G[r,n]-m_r)
//             P_r   = sum_{label==, n!=r} (l - log(exp(l)+s_r)), l=G[r,n]-m_r
//             loss  = mean_r ( -P_r / count_r )
//
// 3 sweeps of the Gram matrix, recomputed via V_WMMA_F32_16X16X4_F32
// (full f32 precision; the 2 MB input stays resident in L2).
// ---------------------------------------------------------------------------

typedef __attribute__((ext_vector_type(2))) float v2f;
typedef __attribute__((ext_vector_type(8))) float v8f;

#define ANCHOR_NUM   1024
#define N_VIEW       8
#define FEAT_DIM     64
#define NROWS        (ANCHOR_NUM * N_VIEW)   // 8192
#define INV_T        10.0f                   // 1 / TEMPERATURE
#define LOSS_SCALE   1.0f                    // TEMPERATURE / BASE_TEMPERATURE
#define WAVES_PB     8
#define BLOCK_T      256
#define COLS_PW      (NROWS / WAVES_PB)      // 1024 columns per wave
#define ROW_TILE     16
#define NBLK         (NROWS / ROW_TILE)      // 512 workgroups
#define KSLICES      (FEAT_DIM / 4)          // 16 chained K=4 WMMA steps

__global__ __launch_bounds__(BLOCK_T)
void pcl_main_kernel(const float* __restrict__ feat,
                     const int*   __restrict__ labels,
                     float*       __restrict__ partial)
{
    __shared__ int   s_lab[ANCHOR_NUM];        // 4 KB label cache
    __shared__ float s_red [WAVES_PB][16];
    __shared__ float s_red2[WAVES_PB][16];
    __shared__ float s_rmax[16];
    __shared__ float s_nsum[16];
    __shared__ float s_loss[16];

    const int tid  = threadIdx.x;
    const int wave = tid >> 5;
    const int lane = tid & 31;
    const int half = lane >> 4;                // 0: rows p, 1: rows p+8
    const int l15  = lane & 15;                // column-within-tile / A row
    const int r0   = blockIdx.x * ROW_TILE;

    for (int i = tid; i < ANCHOR_NUM; i += BLOCK_T) s_lab[i] = labels[i];
    __syncthreads();

    // ---- A operand: 16 rows x 64 K, as 16 K-slices of v2f per lane -------
    // A 16x4 layout: lane<16 holds {K=4k, 4k+1} of row l15, lane>=16 holds
    // {K=4k+2, 4k+3} of the same row -> per-lane float2 at offset 2*half.
    v2f a[KSLICES];
    {
        const float* arow = feat + (size_t)(r0 + l15) * FEAT_DIM + 2 * half;
        #pragma unroll
        for (int ks = 0; ks < KSLICES; ++ks)
            a[ks] = *reinterpret_cast<const v2f*>(arow + 4 * ks);
    }

    int rowlab[8], rowidx[8];
    #pragma unroll
    for (int p = 0; p < 8; ++p) {
        rowidx[p] = r0 + p + 8 * half;          // C/D layout: VGPR p, half
        rowlab[p] = s_lab[rowidx[p] >> 3];      // >> log2(N_VIEW)
    }

    const int colbeg = wave * COLS_PW;
    const int colend = colbeg + COLS_PW;

    // 16x16 Gram tile vs columns [cb, cb+16), K=64 via 16 chained f32 WMMAs.
    // B 4x16 layout mirrors A with N<->lane: identical per-lane float2 loads.
    auto tile = [&](int cb) -> v8f {
        const float* brow = feat + (size_t)(cb + l15) * FEAT_DIM + 2 * half;
        v8f c = {};
        #pragma unroll
        for (int ks = 0; ks < KSLICES; ++ks) {
            v2f b = *reinterpret_cast<const v2f*>(brow + 4 * ks);
            c = __builtin_amdgcn_wmma_f32_16x16x4_f32(false, a[ks], false, b,
                                                      (short)0, c, false, false);
        }
        return c;
    };

    // =============== pass 1: row max (includes self) ======================
    float pmax[8];
    #pragma unroll
    for (int p = 0; p < 8; ++p) pmax[p] = -3.0e38f;

    for (int cb = colbeg; cb < colend; cb += 16) {
        if (cb + 16 < colend)
            __builtin_prefetch(feat + (size_t)(cb + 16 + l15) * FEAT_DIM + 32 * half, 0, 0);
        v8f c = tile(cb);
        #pragma unroll
        for (int p = 0; p < 8; ++p) pmax[p] = fmaxf(pmax[p], c[p]);
    }
    #pragma unroll
    for (int p = 0; p < 8; ++p) {
        float v = pmax[p];
        #pragma unroll
        for (int off = 8; off > 0; off >>= 1)
            v = fmaxf(v, __shfl_xor(v, off, 16));
        if (l15 == 0) s_red[wave][p + 8 * half] = v;
    }
    __syncthreads();
    if (tid < 16) {
        float m = s_red[0][tid];
        #pragma unroll
        for (int w = 1; w < WAVES_PB; ++w) m = fmaxf(m, s_red[w][tid]);
        s_rmax[tid] = m * INV_T;                // row max of logits
    }
    __syncthreads();

    float rmax[8];
    #pragma unroll
    for (int p = 0; p < 8; ++p) rmax[p] = s_rmax[p + 8 * half];

    // =============== pass 2: negative exp-sum =============================
    float pns[8];
    #pragma unroll
    for (int p = 0; p < 8; ++p) pns[p] = 0.0f;

    for (int cb = colbeg; cb < colend; cb += 16) {
        if (cb + 16 < colend)
            __builtin_prefetch(feat + (size_t)(cb + 16 + l15) * FEAT_DIM + 32 * half, 0, 0);
        v8f c = tile(cb);
        const int clab = s_lab[(cb + l15) >> 3];
        #pragma unroll
        for (int p = 0; p < 8; ++p) {
            float e = expf(c[p] * INV_T - rmax[p]);   // <= 1, no overflow
            pns[p] += (clab != rowlab[p]) ? e : 0.0f;
        }
    }
    #pragma unroll
    for (int p = 0; p < 8; ++p) {
        float v = pns[p];
        #pragma unroll
        for (int off = 8; off > 0; off >>= 1)
            v += __shfl_xor(v, off, 16);
        if (l15 == 0) s_red[wave][p + 8 * half] = v;
    }
    __syncthreads();
    if (tid < 16) {
        float s = 0.0f;
        #pragma unroll
        for (int w = 0; w < WAVES_PB; ++w) s += s_red[w][tid];
        s_nsum[tid] = s;
    }
    __syncthreads();

    float nsum[8];
    #pragma unroll
    for (int p = 0; p < 8; ++p) nsum[p] = s_nsum[p + 8 * half];

    // =============== pass 3: positive log-prob sum + count ================
    float pps[8], pct[8];
    #pragma unroll
    for (int p = 0; p < 8; ++p) { pps[p] = 0.0f; pct[p] = 0.0f; }

    for (int cb = colbeg; cb < colend; cb += 16) {
        if (cb + 16 < colend)
            __builtin_prefetch(feat + (size_t)(cb + 16 + l15) * FEAT_DIM + 32 * half, 0, 0);
        v8f c = tile(cb);
        const int n    = cb + l15;
        const int clab = s_lab[n >> 3];
        #pragma unroll
        for (int p = 0; p < 8; ++p) {
            if ((clab == rowlab[p]) && (n != rowidx[p])) {
                float l = c[p] * INV_T - rmax[p];
                pps[p] += l - logf(expf(l) + nsum[p]);
                pct[p] += 1.0f;
            }
        }
    }
    #pragma unroll
    for (int p = 0; p < 8; ++p) {
        float v = pps[p], cnt = pct[p];
        #pragma unroll
        for (int off = 8; off > 0; off >>= 1) {
            v   += __shfl_xor(v,   off, 16);
            cnt += __shfl_xor(cnt, off, 16);
        }
        if (l15 == 0) {
            s_red [wave][p + 8 * half] = v;
            s_red2[wave][p + 8 * half] = cnt;
        }
    }
    __syncthreads();
    if (tid < 16) {
        float ps = 0.0f, cn = 0.0f;
        #pragma unroll
        for (int w = 0; w < WAVES_PB; ++w) { ps += s_red[w][tid]; cn += s_red2[w][tid]; }
        s_loss[tid] = -LOSS_SCALE * (ps / cn);  // >=7 positives/row: cn > 0
    }
    __syncthreads();
    if (tid == 0) {
        float acc = 0.0f;
        #pragma unroll
        for (int r = 0; r < 16; ++r) acc += s_loss[r];
        partial[blockIdx.x] = acc;              // deterministic per-block partial
    }
}

// Deterministic final reduction of the 512 per-block partials.
__global__ __launch_bounds__(256)
void pcl_reduce_kernel(const float* __restrict__ partial,
                       float*       __restrict__ out)
{
    __shared__ float s[256];
    const int tid = threadIdx.x;
    s[tid] = partial[tid] + partial[tid + 256];
    __syncthreads();
    #pragma unroll
    for (int off = 128; off > 0; off >>= 1) {
        if (tid < off) s[tid] += s[tid + off];
        __syncthreads();
    }
    if (tid == 0) out[0] = s[0] / (float)NROWS;
}

extern "C" void kernel_launch(void* const* d_in, const int* in_sizes, int n_in,
                              void* d_out, int out_size, void* d_ws, size_t ws_size,
                              hipStream_t stream)
{
    const float* feats  = (const float*)d_in[0];   // [1024, 8, 64] f32, viewed [8192, 64]
    const int*   labels = (const int*)d_in[1];     // [1024] int
    float* partial      = (float*)d_ws;            // 512 floats of scratch
    float* out          = (float*)d_out;           // scalar

    pcl_main_kernel<<<NBLK, BLOCK_T, 0, stream>>>(feats, labels, partial);
    pcl_reduce_kernel<<<1, 256, 0, stream>>>(partial, out);
}